// attention_net_noc_46978352284213
// MI455X (gfx1250) — compile-verified
//
#include <hip/hip_runtime.h>
#include <math.h>

typedef __attribute__((ext_vector_type(2))) float v2f;
typedef __attribute__((ext_vector_type(8))) float v8f;

#define WAVES 8
#define GRIDB 512
#define NEGV  (-1e8f)

// ---- branch-free transcendentals (hardware TRANS ops, no EXEC churn) ----
__device__ inline float fast_tanh(float x) {
#if __has_builtin(__builtin_amdgcn_tanhf)
  return __builtin_amdgcn_tanhf(x);            // v_tanh_f32 (gfx1250)
#else
  // tanh(x) = 1 - 2/(exp2(2*log2e*x) + 1); saturates correctly at +-inf
  float t = __builtin_amdgcn_exp2f(x * 2.8853900817779268f);
  return 1.f - 2.f * __builtin_amdgcn_rcpf(t + 1.f);
#endif
}
__device__ inline float fast_exp(float x) {
  return __builtin_amdgcn_exp2f(x * 1.4426950408889634f);  // v_exp_f32
}

struct Frag { v2f c[4]; };

// Fragment layout (both A-weights and B-data use the same indexing):
// value j (0..1) of chunk k at lane l:  M[row = l&15][4*k + 2*(l>>4) + j]
__device__ inline Frag ld_frag_row(const float* __restrict__ row, int half) {
  Frag f;
#pragma unroll
  for (int k = 0; k < 4; ++k) {
    const float2 p = *reinterpret_cast<const float2*>(row + 4 * k + 2 * half);
    f.c[k] = v2f{p.x, p.y};
  }
  return f;
}

__device__ inline Frag ld_frag_lds(const float* row, int half) {
  Frag f;
#pragma unroll
  for (int k = 0; k < 4; ++k)
    f.c[k] = v2f{row[4 * k + 2 * half], row[4 * k + 2 * half + 1]};
  return f;
}

// full 16x16x16 f32 matmul as 4 chained V_WMMA_F32_16X16X4_F32
__device__ inline v8f wmma16(const Frag& a, const Frag& b, v8f c) {
#pragma unroll
  for (int k = 0; k < 4; ++k)
    c = __builtin_amdgcn_wmma_f32_16x16x4_f32(false, a.c[k], false, b.c[k],
                                              (short)0, c, false, false);
  return c;
}

__device__ inline float xhalf_add(float v) {
  int o = __builtin_amdgcn_ds_swizzle(__float_as_int(v), 0x401F); // xor lane 16
  return v + __int_as_float(o);
}

// One streaming pass: E = tanh(tanh(X W1^T + b1) W2^T + b2); P = E Wr^T;
// logit = tanh(q + P) . V ; then either glimpse accumulation (sumexp, sum w*E)
// or pointer accumulation (sumexp, packed argmax key), with optional mask.
template <bool PTR>
__global__ __launch_bounds__(WAVES * 32) void pass_kernel(
    const float* __restrict__ X, int nTiles,
    const float* __restrict__ W1, const float* __restrict__ b1,
    const float* __restrict__ W2, const float* __restrict__ b2,
    const float* __restrict__ Wr, const float* __restrict__ Vw,
    const float* __restrict__ q,
    float* __restrict__ g_sumexp, float* __restrict__ g_gsum,
    unsigned long long* __restrict__ g_key, const int* __restrict__ mask) {
  __shared__ float tile[WAVES][16][18];  // stride 18: conflict-free, 8B aligned
  __shared__ float s_gsum[16];
  __shared__ float s_sum;
  __shared__ unsigned long long s_key;

  const int lane = threadIdx.x & 31;
  const int wv   = threadIdx.x >> 5;
  const int m    = lane & 15;
  const int half = lane >> 4;

  if (threadIdx.x == 0) { s_sum = 0.f; s_key = 0ull; }
  if (threadIdx.x < 16) s_gsum[threadIdx.x] = 0.f;
  __syncthreads();

  const Frag aW1 = ld_frag_row(W1 + m * 16, half);
  const Frag aW2 = ld_frag_row(W2 + m * 16, half);
  const Frag aWr = ld_frag_row(Wr + m * 16, half);
  v8f b1v, b2v;
  float qv[8], vv[8];
#pragma unroll
  for (int r = 0; r < 8; ++r) {
    b1v[r] = b1[r + 8 * half];
    b2v[r] = b2[r + 8 * half];
    qv[r]  = q[r + 8 * half];
    vv[r]  = Vw[r + 8 * half];
  }

  float* lrow = &tile[wv][m][0];
  for (int t = blockIdx.x * WAVES + wv; t < nTiles; t += gridDim.x * WAVES) {
    const int row = t * 16 + m;
    Frag xb = ld_frag_row(X + (size_t)row * 16, half);

    v8f h = wmma16(aW1, xb, b1v);
#pragma unroll
    for (int r = 0; r < 8; ++r) h[r] = fast_tanh(h[r]);
    // D-layout -> B-fragment via per-wave LDS roundtrip
#pragma unroll
    for (int r = 0; r < 8; r += 2)
      *reinterpret_cast<float2*>(lrow + 8 * half + r) = make_float2(h[r], h[r + 1]);
    asm volatile("s_wait_dscnt 0x0" ::: "memory");
    Frag hb = ld_frag_lds(lrow, half);

    v8f e = wmma16(aW2, hb, b2v);
#pragma unroll
    for (int r = 0; r < 8; ++r) e[r] = fast_tanh(e[r]);
#pragma unroll
    for (int r = 0; r < 8; r += 2)
      *reinterpret_cast<float2*>(lrow + 8 * half + r) = make_float2(e[r], e[r + 1]);
    asm volatile("s_wait_dscnt 0x0" ::: "memory");
    Frag eb = ld_frag_lds(lrow, half);

    v8f zero = {};
    v8f p = wmma16(aWr, eb, zero);
    float part = 0.f;
#pragma unroll
    for (int r = 0; r < 8; ++r) part += fast_tanh(qv[r] + p[r]) * vv[r];
    const float logit = xhalf_add(part);  // full dot across the lane pair

    if (PTR) {
      float lg = logit;
      if (mask) { if (mask[row] == 0) lg = NEGV; }
      if (half == 0) {
        atomicAdd(&s_sum, fast_exp(lg));  // exp(-1e8) -> 0 exactly
        unsigned b = __float_as_uint(lg);
        b ^= (b & 0x80000000u) ? 0xFFFFFFFFu : 0x80000000u;  // order-preserving
        unsigned long long key =
            ((unsigned long long)b << 32) | (unsigned)(~row);
        atomicMax(&s_key, key);
      }
    } else {
      const float w = fast_exp(logit);  // |logit| <= sum|V| ~ 0.7: no max-shift
      if (half == 0) atomicAdd(&s_sum, w);
#pragma unroll
      for (int r = 0; r < 8; ++r) atomicAdd(&s_gsum[8 * half + r], w * e[r]);
    }
  }
  __syncthreads();
  if (PTR) {
    if (threadIdx.x == 0) { atomicAdd(g_sumexp, s_sum); atomicMax(g_key, s_key); }
  } else {
    if (threadIdx.x == 0) atomicAdd(g_sumexp, s_sum);
    if (threadIdx.x < 16) atomicAdd(&g_gsum[threadIdx.x], s_gsum[threadIdx.x]);
  }
}

// ---- tiny glue kernels (trivial 16-dim algebra) ----
// ws layout (floats): 0 qa_j1[16] | 16 qa_m1[16] | 32 qa_j2[16] | 48 qa_m2[16]
// 64 qp_j[16] | 80 qp_m[16] | 96 e_js[16]
// 112 sum1j + 113 gsum1j[16] | 130 sum1m + 131 gsum1m[16]
// 148 sum2j + 149 gsum2j[16] | 166 sum2m + 167 gsum2m[16]
// 184 sumexp_jl | 185 sumexp_ml | 186 key_j(u64) | 188 key_m(u64)

__global__ void k_setup(const float* lj, const float* ajWq, const float* ajbq,
                        const float* amWq, const float* ambq, float* ws) {
  if (threadIdx.x) return;
  for (int i = 112; i < 192; ++i) ws[i] = 0.f;  // zero all accumulators + keys
  for (int i = 0; i < 16; ++i) {
    float s1 = ajbq[i], s2 = ambq[i];
    for (int k = 0; k < 16; ++k) {
      s1 += lj[k] * ajWq[i * 16 + k];
      s2 += lj[k] * amWq[i * 16 + k];
    }
    ws[i] = s1; ws[16 + i] = s2;
  }
}

__global__ void k_mid1(const float* lj, const float* g1W, const float* g1b,
                       const float* jaWq, const float* jabq, float* ws) {
  if (threadIdx.x) return;
  float cat[48];
  const float ij = 1.f / ws[112], im = 1.f / ws[130];
  for (int f = 0; f < 16; ++f) {
    cat[f] = lj[f]; cat[16 + f] = ws[113 + f] * ij; cat[32 + f] = ws[131 + f] * im;
  }
  float g1[16];
  for (int i = 0; i < 16; ++i) {
    float s = g1b[i];
    for (int k = 0; k < 48; ++k) s += cat[k] * g1W[i * 48 + k];
    g1[i] = fast_tanh(s);
  }
  for (int i = 0; i < 16; ++i) {
    float s = jabq[i];
    for (int k = 0; k < 16; ++k) s += g1[k] * jaWq[i * 16 + k];
    ws[64 + i] = s;
  }
}

__global__ void k_mid2(const float* jobs, const float* jW1, const float* jb1,
                       const float* jW2, const float* jb2, const float* ajWq,
                       const float* ajbq, const float* amWq, const float* ambq,
                       float* ws) {
  if (threadIdx.x) return;
  const unsigned long long key = *reinterpret_cast<unsigned long long*>(ws + 186);
  const int sel = (int)(~(unsigned)(key & 0xFFFFFFFFull));
  const float* xr = jobs + (size_t)sel * 16;
  float h[16], e[16];
  for (int i = 0; i < 16; ++i) {
    float s = jb1[i];
    for (int k = 0; k < 16; ++k) s += xr[k] * jW1[i * 16 + k];
    h[i] = fast_tanh(s);
  }
  for (int i = 0; i < 16; ++i) {
    float s = jb2[i];
    for (int k = 0; k < 16; ++k) s += h[k] * jW2[i * 16 + k];
    e[i] = fast_tanh(s);
    ws[96 + i] = e[i];
  }
  for (int i = 0; i < 16; ++i) {
    float s1 = ajbq[i], s2 = ambq[i];
    for (int k = 0; k < 16; ++k) {
      s1 += e[k] * ajWq[i * 16 + k];
      s2 += e[k] * amWq[i * 16 + k];
    }
    ws[32 + i] = s1; ws[48 + i] = s2;
  }
}

__global__ void k_mid3(const float* g2W, const float* g2b, const float* maWq,
                       const float* mabq, float* ws) {
  if (threadIdx.x) return;
  float cat[48];
  const float ij = 1.f / ws[148], im = 1.f / ws[166];
  for (int f = 0; f < 16; ++f) {
    cat[f] = ws[96 + f]; cat[16 + f] = ws[149 + f] * ij; cat[32 + f] = ws[167 + f] * im;
  }
  float g2[16];
  for (int i = 0; i < 16; ++i) {
    float s = g2b[i];
    for (int k = 0; k < 48; ++k) s += cat[k] * g2W[i * 48 + k];
    g2[i] = fast_tanh(s);
  }
  for (int i = 0; i < 16; ++i) {
    float s = mabq[i];
    for (int k = 0; k < 16; ++k) s += g2[k] * maWq[i * 16 + k];
    ws[80 + i] = s;
  }
}

__global__ void k_final(const float* ws, float* out) {
  if (threadIdx.x) return;
  const unsigned long long kj = *reinterpret_cast<const unsigned long long*>(ws + 186);
  const unsigned long long km = *reinterpret_cast<const unsigned long long*>(ws + 188);
  int sj, sm; float lj, lm;
  {
    sj = (int)(~(unsigned)(kj & 0xFFFFFFFFull));
    unsigned ob = (unsigned)(kj >> 32);
    lj = __uint_as_float((ob & 0x80000000u) ? (ob ^ 0x80000000u) : ~ob);
  }
  {
    sm = (int)(~(unsigned)(km & 0xFFFFFFFFull));
    unsigned ob = (unsigned)(km >> 32);
    lm = __uint_as_float((ob & 0x80000000u) ? (ob ^ 0x80000000u) : ~ob);
  }
  out[0] = (float)sj;
  out[1] = (float)sm;
  out[2] = (lj - logf(ws[184])) + (lm - logf(ws[185]));
}

extern "C" void kernel_launch(void* const* d_in, const int* in_sizes, int n_in,
                              void* d_out, int out_size, void* d_ws,
                              size_t ws_size, hipStream_t stream) {
  const float* jobs = (const float*)d_in[0];
  const float* mach = (const float*)d_in[1];
  const int* mask = (const int*)d_in[2];
  const float *jW1 = (const float*)d_in[3], *jb1 = (const float*)d_in[4],
              *jW2 = (const float*)d_in[5], *jb2 = (const float*)d_in[6];
  const float *mW1 = (const float*)d_in[7], *mb1 = (const float*)d_in[8],
              *mW2 = (const float*)d_in[9], *mb2 = (const float*)d_in[10];
  const float *ajWq = (const float*)d_in[11], *ajbq = (const float*)d_in[12],
              *ajWr = (const float*)d_in[13], *ajV = (const float*)d_in[14];
  const float *amWq = (const float*)d_in[15], *ambq = (const float*)d_in[16],
              *amWr = (const float*)d_in[17], *amV = (const float*)d_in[18];
  const float *jaWq = (const float*)d_in[19], *jabq = (const float*)d_in[20],
              *jaWr = (const float*)d_in[21], *jaV = (const float*)d_in[22];
  const float *maWq = (const float*)d_in[23], *mabq = (const float*)d_in[24],
              *maWr = (const float*)d_in[25], *maV = (const float*)d_in[26];
  const float *g1W = (const float*)d_in[27], *g1b = (const float*)d_in[28];
  const float *g2W = (const float*)d_in[29], *g2b = (const float*)d_in[30];
  const float* lastj = (const float*)d_in[31];

  float* ws = (float*)d_ws;
  const int tJ = (in_sizes[0] / 16) / 16;  // 62500 job tiles (exact)
  const int tM = (in_sizes[1] / 16) / 16;  // 31250 machine tiles (exact)
  dim3 grid(GRIDB), blk(WAVES * 32);

  k_setup<<<1, 32, 0, stream>>>(lastj, ajWq, ajbq, amWq, ambq, ws);
  // glimpse 1 (query = last_j)
  pass_kernel<false><<<grid, blk, 0, stream>>>(jobs, tJ, jW1, jb1, jW2, jb2,
      ajWr, ajV, ws + 0, ws + 112, ws + 113, nullptr, nullptr);
  pass_kernel<false><<<grid, blk, 0, stream>>>(mach, tM, mW1, mb1, mW2, mb2,
      amWr, amV, ws + 16, ws + 130, ws + 131, nullptr, nullptr);
  k_mid1<<<1, 32, 0, stream>>>(lastj, g1W, g1b, jaWq, jabq, ws);
  // job pointer (masked argmax + log-softmax denom)
  pass_kernel<true><<<grid, blk, 0, stream>>>(jobs, tJ, jW1, jb1, jW2, jb2,
      jaWr, jaV, ws + 64, ws + 184, nullptr,
      (unsigned long long*)(ws + 186), mask);
  k_mid2<<<1, 32, 0, stream>>>(jobs, jW1, jb1, jW2, jb2, ajWq, ajbq, amWq, ambq, ws);
  // glimpse 2 (query = selected job embedding)
  pass_kernel<false><<<grid, blk, 0, stream>>>(jobs, tJ, jW1, jb1, jW2, jb2,
      ajWr, ajV, ws + 32, ws + 148, ws + 149, nullptr, nullptr);
  pass_kernel<false><<<grid, blk, 0, stream>>>(mach, tM, mW1, mb1, mW2, mb2,
      amWr, amV, ws + 48, ws + 166, ws + 167, nullptr, nullptr);
  k_mid3<<<1, 32, 0, stream>>>(g2W, g2b, maWq, mabq, ws);
  // machine pointer
  pass_kernel<true><<<grid, blk, 0, stream>>>(mach, tM, mW1, mb1, mW2, mb2,
      maWr, maV, ws + 80, ws + 185, nullptr,
      (unsigned long long*)(ws + 188), nullptr);
  k_final<<<1, 32, 0, stream>>>(ws, (float*)d_out);
}